// Sent_Posit_Drmm_Modeler_14963666059461
// MI455X (gfx1250) — compile-verified
//
#include <hip/hip_runtime.h>
#include <hip/hip_bf16.h>
#include <math.h>

#define EMB 30
#define EMBP 32            // padded row stride for WMMA K=32
#define QN 2048
#define DN 8192
#define NTILE (DN / 16)    // 512 doc tiles
#define NEG_INF (-1.0e30f)
#define OH_THRESH 0.999f

typedef __attribute__((ext_vector_type(2))) float v2f;
typedef __attribute__((ext_vector_type(8))) float v8f;

// ---------------------------------------------------------------------------
// Kernel 1: trigram conv (SAME, pad 1) + leaky + residual, fused with row
// inverse L2 norms. Rewrites BOTH the raw row and the conv row into
// workspace with stride-32 zero-padded layout so the WMMA kernel can use
// unconditional aligned b64 fragment loads. One wave per row.
// ---------------------------------------------------------------------------
__global__ __launch_bounds__(32) void conv_norm_kernel(
    const float* __restrict__ x, int L,
    const float* __restrict__ W,      // [EMB_out, EMB_in, 3] row-major
    const float* __restrict__ b,      // [EMB]
    float* __restrict__ xpad,         // [L, 32] padded raw copy
    float* __restrict__ ypad,         // [L, 32] padded conv output
    float* __restrict__ inv_raw,      // [L]
    float* __restrict__ inv_conv) {   // [L]
  __shared__ float xl[90];            // rows i-1, i, i+1
  __shared__ float red[64];
  const int i = blockIdx.x;
  const int t = threadIdx.x;

  for (int j = t; j < 90; j += 32) {
    int r = i - 1 + j / EMB;
    int c = j % EMB;
    xl[j] = (r >= 0 && r < L) ? x[r * EMB + c] : 0.0f;
  }
  __syncthreads();

  float outv = 0.0f, xv = 0.0f;
  if (t < EMB) {
    float acc = b[t];
    #pragma unroll
    for (int k = 0; k < 3; ++k)
      #pragma unroll
      for (int ci = 0; ci < EMB; ++ci)
        acc = fmaf(xl[k * EMB + ci], W[t * 90 + ci * 3 + k], acc);
    float lv = acc > 0.0f ? acc : 0.1f * acc;   // leaky
    xv = xl[EMB + t];
    outv = lv + xv;                              // residual
  }
  // padded stride-32 stores; lanes 30,31 write the K-padding zeros
  xpad[i * EMBP + t] = xv;
  ypad[i * EMBP + t] = outv;

  red[t] = xv * xv;
  red[32 + t] = outv * outv;
  __syncthreads();
  for (int s = 16; s > 0; s >>= 1) {
    if (t < s) { red[t] += red[t + s]; red[32 + t] += red[32 + t + s]; }
    __syncthreads();
  }
  if (t == 0) {
    inv_raw[i]  = rsqrtf(red[0]);
    inv_conv[i] = rsqrtf(red[32]);
  }
}

// ---------------------------------------------------------------------------
// Doc-tile register buffer: 16 v2f WMMA B-fragments (raw + conv) + the two
// inverse-norm scalars for this tile's 16 doc rows (N = l15 per lane).
// ---------------------------------------------------------------------------
struct BTile {
  v2f r[8], c[8];
  float ibr, ibc;
};

__device__ __forceinline__ void load_tile(
    BTile& t, const float* __restrict__ Br, const float* __restrict__ Bc,
    const float* __restrict__ invBr, const float* __restrict__ invBc,
    int dbase, int l15, int half) {
  const float* pbr = Br + (dbase + l15) * EMBP;
  const float* pbc = Bc + (dbase + l15) * EMBP;
  #pragma unroll
  for (int ks = 0; ks < 8; ++ks) {
    int k0 = ks * 4 + half * 2;
    t.r[ks] = *(const v2f*)(pbr + k0);
    t.c[ks] = *(const v2f*)(pbc + k0);
  }
  t.ibr = invBr[dbase + l15];
  t.ibc = invBc[dbase + l15];
}

// ---------------------------------------------------------------------------
// Kernel 2: fused cosine-sim WMMA GEMM + top-5 pooling, both docs
// (blockIdx.y selects doc). Block = 128 threads (4 waves), owns 16 query
// rows; each wave streams 16-wide doc tiles with REGISTER DOUBLE BUFFERING:
// while tile t runs its 16x V_WMMA_F32_16X16X4_F32 chain + epilogue + top-k,
// tile t+4's fragments are already in flight into the other buffer.
// Lanes 0-15 keep top5 + one-hot count of sim_i rows; lanes 16-31 keep top5
// of sim_s rows.
// ---------------------------------------------------------------------------
__global__ __launch_bounds__(128) void sim_pool_kernel(
    const float* __restrict__ Ar, const float* __restrict__ Ac,   // [QN,32]
    const float* __restrict__ invAr, const float* __restrict__ invAc,
    const float* __restrict__ Br1, const float* __restrict__ Bc1, // [DN,32]
    const float* __restrict__ iBr1, const float* __restrict__ iBc1,
    const float* __restrict__ Br2, const float* __restrict__ Bc2,
    const float* __restrict__ iBr2, const float* __restrict__ iBc2,
    float* __restrict__ pools /* [2,QN,6] */) {
  __shared__ float simbuf[4][512];   // per wave: [0..255] sim_i, [256..511] sim_s
  __shared__ float mtop[4][32][5];
  __shared__ int   mcnt[4][16];

  const int tid  = threadIdx.x;
  const int wave = tid >> 5;
  const int lane = tid & 31;
  const int half = lane >> 4;
  const int l15  = lane & 15;
  const int qbase = blockIdx.x * 16;
  const int doc   = blockIdx.y;

  const float* __restrict__ Br    = doc ? Br2  : Br1;
  const float* __restrict__ Bc    = doc ? Bc2  : Bc1;
  const float* __restrict__ invBr = doc ? iBr2 : iBr1;
  const float* __restrict__ invBc = doc ? iBc2 : iBc1;
  float* __restrict__ dpools = pools + doc * QN * 6;

  // --- A fragments (loop invariant): 16x4 f32 layout, K split over halves ---
  v2f aR[8], aC[8];
  {
    const float* pr = Ar + (qbase + l15) * EMBP;
    const float* pc = Ac + (qbase + l15) * EMBP;
    #pragma unroll
    for (int ks = 0; ks < 8; ++ks) {
      int k0 = ks * 4 + half * 2;
      aR[ks] = *(const v2f*)(pr + k0);
      aC[ks] = *(const v2f*)(pc + k0);
    }
  }
  // invA per C-register row: C vgpr r holds M = r + half*8
  float iAr[8], iAc[8];
  #pragma unroll
  for (int r = 0; r < 8; ++r) {
    int m = r + half * 8;
    iAr[r] = invAr[qbase + m];
    iAc[r] = invAc[qbase + m];
  }

  float t0 = NEG_INF, t1 = NEG_INF, t2 = NEG_INF, t3 = NEG_INF, t4 = NEG_INF;
  int cnt = 0;

  // consume one tile: WMMA chain -> cosine scaling -> LDS stage -> top-k scan
  auto process = [&](const BTile& bt) {
    v8f cR = {}; v8f cC = {};
    #pragma unroll
    for (int ks = 0; ks < 8; ++ks) {
      cR = __builtin_amdgcn_wmma_f32_16x16x4_f32(false, aR[ks], false, bt.r[ks],
                                                 (short)0, cR, false, false);
      cC = __builtin_amdgcn_wmma_f32_16x16x4_f32(false, aC[ks], false, bt.c[ks],
                                                 (short)0, cC, false, false);
    }
    #pragma unroll
    for (int r = 0; r < 8; ++r) {
      int m = r + half * 8;
      simbuf[wave][m * 16 + l15]       = cR[r] * iAr[r] * bt.ibr;
      simbuf[wave][256 + m * 16 + l15] = cC[r] * iAc[r] * bt.ibc;
    }
    __syncthreads();
    const float* row = (lane < 16) ? &simbuf[wave][lane * 16]
                                   : &simbuf[wave][256 + (lane - 16) * 16];
    #pragma unroll
    for (int n = 0; n < 16; ++n) {
      float v = row[n];
      if (lane < 16) cnt += (v > OH_THRESH) ? 1 : 0;
      if (v > t4) {
        if      (v > t0) { t4 = t3; t3 = t2; t2 = t1; t1 = t0; t0 = v; }
        else if (v > t1) { t4 = t3; t3 = t2; t2 = t1; t1 = v; }
        else if (v > t2) { t4 = t3; t3 = t2; t2 = v; }
        else if (v > t3) { t4 = t3; t3 = v; }
        else             { t4 = v; }
      }
    }
    __syncthreads();
  };

  // software pipeline: 2 tiles per iteration, ping-pong register buffers
  BTile buf0, buf1;
  load_tile(buf0, Br, Bc, invBr, invBc, wave * 16, l15, half);
  for (int dt = wave; dt < NTILE; dt += 8) {
    // L2 prefetch beyond the register pipeline
    if (dt + 12 < NTILE)
      __builtin_prefetch(Br + (dt + 12) * 16 * EMBP + l15 * EMBP, 0, 1);
    load_tile(buf1, Br, Bc, invBr, invBc, (dt + 4) * 16, l15, half);
    process(buf0);
    if (dt + 8 < NTILE)
      load_tile(buf0, Br, Bc, invBr, invBc, (dt + 8) * 16, l15, half);
    process(buf1);
  }

  // cross-wave merge
  mtop[wave][lane][0] = t0; mtop[wave][lane][1] = t1; mtop[wave][lane][2] = t2;
  mtop[wave][lane][3] = t3; mtop[wave][lane][4] = t4;
  if (lane < 16) mcnt[wave][lane] = cnt;
  __syncthreads();

  if (wave == 0) {
    float cand[20];
    #pragma unroll
    for (int w = 0; w < 4; ++w)
      #pragma unroll
      for (int j = 0; j < 5; ++j)
        cand[w * 5 + j] = mtop[w][lane][j];
    float sel[5];
    #pragma unroll
    for (int s = 0; s < 5; ++s) {
      int bi = 0; float bv = cand[0];
      #pragma unroll
      for (int j = 1; j < 20; ++j)
        if (cand[j] > bv) { bv = cand[j]; bi = j; }
      sel[s] = bv; cand[bi] = NEG_INF;
    }
    float tmax  = sel[0];
    float tmean = (sel[0] + sel[1] + sel[2] + sel[3] + sel[4]) * 0.2f;
    if (lane < 16) {
      int c = mcnt[0][lane] + mcnt[1][lane] + mcnt[2][lane] + mcnt[3][lane];
      int c5 = c > 5 ? 5 : c;
      float* p = dpools + (qbase + lane) * 6;
      p[0] = (c > 0) ? 1.0f : 0.0f;   // oh max
      p[1] = (float)c5 * 0.2f;        // oh top-5 mean
      p[2] = tmax;                    // sim_i max
      p[3] = tmean;                   // sim_i top-5 mean
    } else {
      float* p = dpools + (qbase + lane - 16) * 6;
      p[4] = tmax;                    // sim_s max
      p[5] = tmean;                   // sim_s top-5 mean
    }
  }
}

// ---------------------------------------------------------------------------
// Kernel 3: q_weights = softmax( [q_conv, q_idf] @ W_qw.T ). One block.
// qc is the padded stride-32 conv layout.
// ---------------------------------------------------------------------------
__global__ __launch_bounds__(256) void qweight_kernel(
    const float* __restrict__ qc, const float* __restrict__ qidf,
    const float* __restrict__ Wqw, float* __restrict__ logits,
    float* __restrict__ wts) {
  __shared__ float red[256];
  __shared__ float s_max, s_sum;
  const int t = threadIdx.x;
  float lmax = NEG_INF;
  for (int q = t; q < QN; q += 256) {
    float acc = 0.0f;
    const float* p = qc + q * EMBP;
    #pragma unroll
    for (int e = 0; e < EMB; ++e) acc = fmaf(p[e], Wqw[e], acc);
    acc = fmaf(qidf[q], Wqw[EMB], acc);
    logits[q] = acc;
    lmax = fmaxf(lmax, acc);
  }
  red[t] = lmax; __syncthreads();
  for (int s = 128; s > 0; s >>= 1) {
    if (t < s) red[t] = fmaxf(red[t], red[t + s]);
    __syncthreads();
  }
  if (t == 0) s_max = red[0];
  __syncthreads();
  const float m = s_max;
  float lsum = 0.0f;
  for (int q = t; q < QN; q += 256) lsum += __expf(logits[q] - m);
  red[t] = lsum; __syncthreads();
  for (int s = 128; s > 0; s >>= 1) {
    if (t < s) red[t] += red[t + s];
    __syncthreads();
  }
  if (t == 0) s_sum = red[0];
  __syncthreads();
  const float inv = 1.0f / s_sum;
  for (int q = t; q < QN; q += 256) wts[q] = __expf(logits[q] - m) * inv;
}

// ---------------------------------------------------------------------------
// Kernel 4: pool6 -> 8 (leaky) -> 1 MLP, weighted mean per doc, hinge loss.
// ---------------------------------------------------------------------------
__global__ __launch_bounds__(256) void final_kernel(
    const float* __restrict__ pools,   // [2, QN, 6]
    const float* __restrict__ wts,
    const float* __restrict__ W1, const float* __restrict__ W2,
    const float* __restrict__ gaf, const float* __restrict__ baf,
    const float* __restrict__ Wout, float* __restrict__ out) {
  __shared__ float sW1[48], sW2[8];
  __shared__ float red[512];
  const int t = threadIdx.x;
  if (t < 48) sW1[t] = W1[t];
  if (t < 8)  sW2[t] = W2[t];
  __syncthreads();
  float acc0 = 0.0f, acc1 = 0.0f;
  for (int q = t; q < QN; q += 256) {
    const float wq = wts[q];
    #pragma unroll
    for (int d = 0; d < 2; ++d) {
      const float* p = pools + (d * QN + q) * 6;
      float lo = 0.0f;
      #pragma unroll
      for (int j = 0; j < 8; ++j) {
        float h = 0.0f;
        #pragma unroll
        for (int k = 0; k < 6; ++k) h = fmaf(p[k], sW1[j * 6 + k], h);
        h = h > 0.0f ? h : 0.1f * h;   // leaky
        lo = fmaf(h, sW2[j], lo);
      }
      if (d == 0) acc0 += lo * wq; else acc1 += lo * wq;
    }
  }
  red[t] = acc0; red[256 + t] = acc1;
  __syncthreads();
  for (int s = 128; s > 0; s >>= 1) {
    if (t < s) { red[t] += red[t + s]; red[256 + t] += red[256 + t + s]; }
    __syncthreads();
  }
  if (t == 0) {
    float e1 = red[0]   / (float)QN;   // doc1_emit
    float e2 = red[256] / (float)QN;   // doc2_emit
    float good = e1 * Wout[4], bad = e2 * Wout[4];
    #pragma unroll
    for (int i = 0; i < 4; ++i) {
      good = fmaf(gaf[i], Wout[i], good);
      bad  = fmaf(baf[i], Wout[i], bad);
    }
    float loss = 1.0f + bad - good;
    loss = loss > 0.0f ? loss : 0.0f;
    out[0] = loss; out[1] = good; out[2] = bad;
  }
}

// ---------------------------------------------------------------------------
extern "C" void kernel_launch(void* const* d_in, const int* in_sizes, int n_in,
                              void* d_out, int out_size, void* d_ws, size_t ws_size,
                              hipStream_t stream) {
  const float* doc1 = (const float*)d_in[0];
  const float* doc2 = (const float*)d_in[1];
  const float* ques = (const float*)d_in[2];
  const float* qidf = (const float*)d_in[3];
  const float* gaf  = (const float*)d_in[4];
  const float* baf  = (const float*)d_in[5];
  const float* Wc   = (const float*)d_in[6];
  const float* bc   = (const float*)d_in[7];
  const float* Wqw  = (const float*)d_in[8];
  const float* Wq1  = (const float*)d_in[9];
  const float* Wq2  = (const float*)d_in[10];
  const float* Wout = (const float*)d_in[11];
  float* out = (float*)d_out;

  float* ws = (float*)d_ws;
  float* qp     = ws; ws += QN * EMBP;   // padded raw q
  float* qc     = ws; ws += QN * EMBP;   // padded conv q
  float* d1p    = ws; ws += DN * EMBP;
  float* d1c    = ws; ws += DN * EMBP;
  float* d2p    = ws; ws += DN * EMBP;
  float* d2c    = ws; ws += DN * EMBP;
  float* iqr    = ws; ws += QN;
  float* iqc    = ws; ws += QN;
  float* id1r   = ws; ws += DN;
  float* id1c   = ws; ws += DN;
  float* id2r   = ws; ws += DN;
  float* id2c   = ws; ws += DN;
  float* pools  = ws; ws += 2 * QN * 6;
  float* logits = ws; ws += QN;
  float* wts    = ws; ws += QN;

  // 1) conv + residual + inverse norms + padded re-layout
  conv_norm_kernel<<<QN, 32, 0, stream>>>(ques, QN, Wc, bc, qp,  qc,  iqr,  iqc);
  conv_norm_kernel<<<DN, 32, 0, stream>>>(doc1, DN, Wc, bc, d1p, d1c, id1r, id1c);
  conv_norm_kernel<<<DN, 32, 0, stream>>>(doc2, DN, Wc, bc, d2p, d2c, id2r, id2c);

  // 2) fused WMMA cosine-sim + top-5 pools, both docs in one launch
  dim3 grid(QN / 16, 2);
  sim_pool_kernel<<<grid, 128, 0, stream>>>(
      qp, qc, iqr, iqc,
      d1p, d1c, id1r, id1c,
      d2p, d2c, id2r, id2c,
      pools);

  // 3) query softmax weights
  qweight_kernel<<<1, 256, 0, stream>>>(qc, qidf, Wqw, logits, wts);

  // 4) MLP + weighted mean + hinge loss
  final_kernel<<<1, 256, 0, stream>>>(pools, wts, Wq1, Wq2, gaf, baf, Wout, out);
}